// EigenRatioPerPoints_28484223107624
// MI455X (gfx1250) — compile-verified
//
#include <hip/hip_runtime.h>
#include <math.h>

// MI455X / gfx1250, wave32. Compile-only target.

typedef float v2f __attribute__((ext_vector_type(2)));
typedef float v8f __attribute__((ext_vector_type(8)));

#define NB      4
#define NPTS    8192
#define KNN     16
#define CHUNK   2048
#define THREADS 256
#define WAVES   (THREADS / 32)
#define QPB     (WAVES * 16)   // 128 queries per block

__global__ __launch_bounds__(THREADS)
void EigenRatioPerPoints_kernel(const float* __restrict__ xg,
                                float* __restrict__ out) {
    // 32KB LDS: phase 1 = candidate chunk (2048 x float4); phase 2 = merge lists
    __shared__ __align__(16) unsigned char smem[32 * 1024];
    float4* chunk  = (float4*)smem;
    // Same bytes viewed as float2 pairs: point p = { (x,y) , (z,|p|^2) }
    const float2* chunk2 = (const float2*)smem;

    const int tid    = threadIdx.x;
    const int wave   = tid >> 5;
    const int lane   = tid & 31;
    const int lane16 = lane & 15;
    const bool lo    = (lane < 16);

    const int b      = blockIdx.x / (NPTS / QPB);
    const int qBlock = (blockIdx.x % (NPTS / QPB)) * QPB;
    const int q      = qBlock + wave * 16 + lane16;   // this lane's query point

    const float* __restrict__ xb = xg + (size_t)b * NPTS * 3;

    // ---- B matrix (K=4 x N=16): query columns, homogeneous bias trick ----
    // col(q) = (-2qx, -2qy, -2qz, 1)  ->  D[m][n] = |c_m|^2 - 2 c_m.q_n = d2 - |q_n|^2
    const float qx = xb[q * 3 + 0];
    const float qy = xb[q * 3 + 1];
    const float qz = xb[q * 3 + 2];
    v2f bq;
    bq[0] = lo ? (-2.0f * qx) : (-2.0f * qz);
    bq[1] = lo ? (-2.0f * qy) : 1.0f;

    // A-operand LDS fetch offset (in float2 units), fixed per lane:
    // lanes 0-15 take (x,y) of point lane16; lanes 16-31 take (z,|p|^2)
    const int laneOff = lane16 * 2 + (lo ? 0 : 1);

    // ---- register-resident sorted top-16 (ascending), in D-space ----
    float kd[KNN];
    int   ki[KNN];
#pragma unroll
    for (int j = 0; j < KNN; ++j) { kd[j] = 3.4e38f; ki[j] = 0; }

    const v8f czero = {};

    // One 16-candidate WMMA tile + streaming top-k scan
    auto processTile = [&](int c0, int t, float2 f2) {
        v2f a; a[0] = f2.x; a[1] = f2.y;
        v8f acc = __builtin_amdgcn_wmma_f32_16x16x4_f32(
            false, a, false, bq, (short)0, czero, false, false);
        // lane<16: rows v (v=0..7), lane>=16: rows v+8; col = lane%16 = our query
        const int base = c0 + t * 16 + (lo ? 0 : 8);
#pragma unroll
        for (int v = 0; v < 8; ++v) {
            const float d = acc[v];
            if (d < kd[KNN - 1]) {          // rare: streaming top-k insert
                float cd = d;
                int   ci = base + v;
#pragma unroll
                for (int j = 0; j < KNN; ++j) {
                    const bool  sm = cd < kd[j];
                    const float od = kd[j];
                    const int   oi = ki[j];
                    kd[j] = sm ? cd : od;
                    ki[j] = sm ? ci : oi;
                    cd    = sm ? od : cd;
                    ci    = sm ? oi : ci;
                }
            }
        }
    };

    for (int c0 = 0; c0 < NPTS; c0 += CHUNK) {
        __syncthreads();
        // Stage chunk into LDS as (x,y,z,|p|^2)
        for (int t = tid; t < CHUNK; t += THREADS) {
            const int g = c0 + t;
            const float px = xb[g * 3 + 0];
            const float py = xb[g * 3 + 1];
            const float pz = xb[g * 3 + 2];
            float4 v;
            v.x = px; v.y = py; v.z = pz;
            v.w = px * px + py * py + pz * pz;
            chunk[t] = v;
        }
        if (c0 + CHUNK < NPTS) {
            // global_prefetch_b8 the next chunk (one cacheline probe per thread)
            __builtin_prefetch(xb + (size_t)(c0 + CHUNK) * 3 + (size_t)tid * 24, 0, 1);
        }
        __syncthreads();

        // ---- tile loop, 4x unrolled: batch 4 independent ds_load_b64 +
        // 4 independent WMMAs so LDS latency overlaps matrix work ----
        for (int t = 0; t < CHUNK / 16; t += 4) {
            const float2 f0 = chunk2[(t + 0) * 32 + laneOff];
            const float2 f1 = chunk2[(t + 1) * 32 + laneOff];
            const float2 f2 = chunk2[(t + 2) * 32 + laneOff];
            const float2 f3 = chunk2[(t + 3) * 32 + laneOff];
            processTile(c0, t + 0, f0);
            processTile(c0, t + 1, f1);
            processTile(c0, t + 2, f2);
            processTile(c0, t + 3, f3);
        }
    }

    // ---- merge the lane / lane+16 partial top-16 lists through LDS ----
    __syncthreads();                      // chunk buffer no longer needed
    float* md = (float*)smem;             // 128 queries * 32 floats = 16KB
    int*   mi = (int*)(smem + 16 * 1024); // 128 queries * 32 ints   = 16KB
    const int qLocal = wave * 16 + lane16;
    const int slot   = qLocal * 32 + (lo ? 0 : 16);
#pragma unroll
    for (int j = 0; j < KNN; ++j) { md[slot + j] = kd[j]; mi[slot + j] = ki[j]; }
    asm volatile("s_wait_dscnt 0" ::: "memory");
    __syncthreads();

    if (lo) {
        // two-pointer merge of two sorted 16-lists; only the SET of 16 nearest
        // matters (mean/cov are permutation invariant)
        int ia = 0, ib = 0;
        float sx = 0, sy = 0, sz = 0;
        float sxx = 0, sxy = 0, sxz = 0, syy = 0, syz = 0, szz = 0;
        const int base0 = qLocal * 32;
        for (int t = 0; t < KNN; ++t) {
            const float da = md[base0 + ia];
            const float db = md[base0 + 16 + ib];
            int gi;
            if (da <= db) { gi = mi[base0 + ia];      ++ia; }
            else          { gi = mi[base0 + 16 + ib]; ++ib; }
            const float px = xb[gi * 3 + 0];
            const float py = xb[gi * 3 + 1];
            const float pz = xb[gi * 3 + 2];
            sx += px; sy += py; sz += pz;
            sxx += px * px; sxy += px * py; sxz += px * pz;
            syy += py * py; syz += py * pz; szz += pz * pz;
        }
        const float ik = 1.0f / (float)KNN;
        const float mx = sx * ik, my = sy * ik, mz = sz * ik;
        const float a00 = sxx * ik - mx * mx;
        const float a01 = sxy * ik - mx * my;
        const float a02 = sxz * ik - mx * mz;
        const float a11 = syy * ik - my * my;
        const float a12 = syz * ik - my * mz;
        const float a22 = szz * ik - mz * mz;

        // closed-form symmetric 3x3 eigenvalues (trigonometric method)
        const float p1 = a01 * a01 + a02 * a02 + a12 * a12;
        const float qm = (a00 + a11 + a22) * (1.0f / 3.0f);
        const float d0 = a00 - qm, d1 = a11 - qm, d2 = a22 - qm;
        const float p2 = d0 * d0 + d1 * d1 + d2 * d2 + 2.0f * p1;
        const float p  = sqrtf(p2 * (1.0f / 6.0f));
        float ratio;
        if (p < 1e-30f) {
            ratio = 1.0f;                 // isotropic: all eigenvalues equal
        } else {
            const float ip  = 1.0f / p;
            const float b00 = d0 * ip, b01 = a01 * ip, b02 = a02 * ip;
            const float b11 = d1 * ip, b12 = a12 * ip, b22 = d2 * ip;
            float r = 0.5f * (b00 * (b11 * b22 - b12 * b12)
                            - b01 * (b01 * b22 - b12 * b02)
                            + b02 * (b01 * b12 - b11 * b02));
            r = fminf(1.0f, fmaxf(-1.0f, r));
            const float phi = acosf(r) * (1.0f / 3.0f);
            const float e1  = qm + 2.0f * p * cosf(phi);                       // largest
            const float e3  = qm + 2.0f * p * cosf(phi + 2.0943951023931953f); // smallest
            const float e2  = 3.0f * qm - e1 - e3;                             // middle
            ratio = e1 / e2;
        }
        out[(size_t)b * NPTS + q] = ratio;
    }
}

extern "C" void kernel_launch(void* const* d_in, const int* in_sizes, int n_in,
                              void* d_out, int out_size, void* d_ws, size_t ws_size,
                              hipStream_t stream) {
    (void)in_sizes; (void)n_in; (void)out_size; (void)d_ws; (void)ws_size;
    const float* x = (const float*)d_in[0];   // (4, 8192, 3) float32; d_in[1] = k == 16 (fixed)
    float* out = (float*)d_out;               // (4, 8192) float32
    dim3 grid(NB * (NPTS / QPB));             // 256 blocks
    dim3 block(THREADS);                      // 8 wave32s
    EigenRatioPerPoints_kernel<<<grid, block, 0, stream>>>(x, out);
}